// Decoder_82231443849558
// MI455X (gfx1250) — compile-verified
//
#include <hip/hip_runtime.h>
#include <math.h>

// ---------------------------------------------------------------------------
// SO(3) equivariant decoder on MI455X (gfx1250), fp32 via V_WMMA_F32_16X16X4_F32
// ---------------------------------------------------------------------------

typedef float v2f __attribute__((ext_vector_type(2)));
typedef float v8f __attribute__((ext_vector_type(8)));

#define GAIN 1.41421356237309515f  // sqrt(2) (normalize2mom relu gain)

#define DIM10 1771   // dim_so3(10)
#define DIM15 5456   // dim_so3(15)
#define DIM20 12341  // dim_so3(20)
#define NGRID 1728   // 12^3 SO(3) grid
#define NKER  192    // 8*3*8 kernel grid
#define NYG   24     // 8*3 s2 kernel grid
#define NSPH  441    // (20+1)^2

__device__ __forceinline__ int off_so3_d(int l) { return l * (2 * l - 1) * (2 * l + 1) / 3; }
static inline size_t off_so3_h(int l) { return (size_t)l * (2 * l - 1) * (2 * l + 1) / 3; }

// ------------------------------ WMMA GEMM core -----------------------------
#define TM 128
#define TN 64
#define KC 16
#define LDA_S 17
#define LDB_S 65

__device__ __forceinline__ v8f wmma4(v2f a, v2f b, v8f c) {
  return __builtin_amdgcn_wmma_f32_16x16x4_f32(false, a, false, b, (short)0, c, false, false);
}

template <class AL, class BL, class EP>
__device__ __forceinline__ void gemm_core(int m0, int n0, int Kdim,
                                          const AL& A, const BL& B, const EP& ep) {
  __shared__ float As[TM * LDA_S];
  __shared__ float Bs[KC * LDB_S];
  const int tid = threadIdx.x;
  const int lane = tid & 31, wave = tid >> 5;
  const int wm = wave >> 1, wn = wave & 1;     // 4x2 wave grid -> 128x64 macro tile
  const int lhalf = lane >> 4, lmod = lane & 15;

  v8f acc[2][2] = {};
  const int nch = (Kdim + KC - 1) / KC;
  for (int ch = 0; ch < nch; ++ch) {
    const int kb = ch * KC;
#pragma unroll
    for (int i = 0; i < 8; ++i) {              // stage A tile: 128x16
      int e = tid + i * 256;
      int r = e >> 4, k = e & 15;
      As[r * LDA_S + k] = A(m0 + r, kb + k);
    }
#pragma unroll
    for (int i = 0; i < 4; ++i) {              // stage B tile: 16x64
      int e = tid + i * 256;
      int k = e >> 6, n = e & 63;
      Bs[k * LDB_S + n] = B(kb + k, n0 + n);
    }
    __syncthreads();
#pragma unroll
    for (int kk = 0; kk < KC; kk += 4) {
      v2f a[2], b[2];
#pragma unroll
      for (int ti = 0; ti < 2; ++ti) {
        int mr = wm * 32 + ti * 16 + lmod;
        a[ti].x = As[mr * LDA_S + kk + 2 * lhalf];
        a[ti].y = As[mr * LDA_S + kk + 2 * lhalf + 1];
      }
#pragma unroll
      for (int tj = 0; tj < 2; ++tj) {
        int nc = wn * 32 + tj * 16 + lmod;
        b[tj].x = Bs[(kk + 2 * lhalf) * LDB_S + nc];
        b[tj].y = Bs[(kk + 2 * lhalf + 1) * LDB_S + nc];
      }
#pragma unroll
      for (int ti = 0; ti < 2; ++ti)
#pragma unroll
        for (int tj = 0; tj < 2; ++tj) acc[ti][tj] = wmma4(a[ti], b[tj], acc[ti][tj]);
    }
    __syncthreads();
  }
#pragma unroll
  for (int ti = 0; ti < 2; ++ti)
#pragma unroll
    for (int tj = 0; tj < 2; ++tj)
#pragma unroll
      for (int r = 0; r < 8; ++r) {
        int row = m0 + wm * 32 + ti * 16 + r + 8 * lhalf;
        int col = n0 + wn * 32 + tj * 16 + lmod;
        ep(row, col, acc[ti][tj][r]);
      }
}

// ------------------------------ loaders / epilogues ------------------------
struct LdRM {  // dense row-major with bounds
  const float* p; int ld, R, C;
  __device__ float operator()(int r, int c) const {
    return (r < R && c < C) ? p[(size_t)r * ld + c] : 0.f;
  }
};
struct LdK {  // K slice: row=(g,m), k=(f,u) -> kb[(f*G+g)*ld + off + u*d + m]
  const float* p; int ld, d, G, off, R, Kb;
  __device__ float operator()(int r, int k) const {
    if (r >= R || k >= Kb) return 0.f;
    int g = r / d, m = r - g * d, f = k / d, u = k - f * d;
    return p[(size_t)(f * G + g) * ld + off + u * d + m];
  }
};
struct LdX {  // input coeffs: k=(f,u), col=(b,j) -> x[b, f0+f, off + u*d + j]
  const float* p; int d, off, F, dimIn, f0, Kb, Nb;
  __device__ float operator()(int k, int c) const {
    if (k >= Kb || c >= Nb) return 0.f;
    int f = k / d, u = k - f * d, b = c / d, j = c - b * d;
    return p[((size_t)b * F + f0 + f) * dimIn + off + u * d + j];
  }
};
struct EpStore {
  float* p; int ld, M, N;
  __device__ void operator()(int r, int c, float v) const {
    if (r < M && c < N) p[(size_t)r * ld + c] = v;
  }
};
struct EpScale {
  float* p; int ld, M, N; float s;
  __device__ void operator()(int r, int c, float v) const {
    if (r < M && c < N) p[(size_t)r * ld + c] = v * s;
  }
};
struct EpSynth {  // S2[bf,n] = GAIN*relu(acc)*qw[n]
  float* p; const float* qw; int M;
  __device__ void operator()(int r, int c, float v) const {
    if (r < M && c < NGRID) p[(size_t)r * NGRID + c] = GAIN * fmaxf(v, 0.f) * qw[c];
  }
};
struct EpAcc {  // out[b, g, off + m*d + j] += acc  (accumulates across f-chunk launches)
  float* p; int d, off, G, dimOut, R, Nb;
  __device__ void operator()(int r, int c, float v) const {
    if (r >= R || c >= Nb) return;
    int g = r / d, m = r - g * d, b = c / d, j = c - b * d;
    p[((size_t)b * G + g) * dimOut + off + m * d + j] += v;
  }
};

// ------------------------------ GEMM wrappers ------------------------------
template <class EP>
__global__ void k_gemm_rm(const float* A, int lda, const float* B, int ldb,
                          int M, int N, int K, EP ep) {
  int ntn = (N + TN - 1) / TN, ntm = (M + TM - 1) / TM;
  int t = blockIdx.x;
  if (t >= ntm * ntn) return;
  int tm = t / ntn, tn = t - tm * ntn;
  LdRM a{A, lda, M, K};
  LdRM b{B, ldb, K, N};
  gemm_core(tm * TM, tn * TN, K, a, b, ep);
}

// K-slice build: K[(f_rel*G+g), (u,m)] = scale * sum_n w[f0+f_rel,g,n] Dk_l[n,u,m]
__global__ void k_build(const float* w, int rows, const float* dkf, float* kbuf,
                        int ldk, float scale) {
  int l = blockIdx.y;
  int d = 2 * l + 1, dd = d * d, off = off_so3_d(l);
  int ntn = (dd + TN - 1) / TN, ntm = (rows + TM - 1) / TM;
  int t = blockIdx.x;
  if (t >= ntm * ntn) return;
  int tm = t / ntn, tn = t - tm * ntn;
  LdRM a{w, NKER, rows, NKER};
  LdRM b{dkf + off, DIM15, NKER, dd};
  EpScale ep{kbuf + off, ldk, rows, dd, scale};
  gemm_core(tm * TM, tn * TN, NKER, a, b, ep);
}

// conv apply: out[b,g,m,j] += sum_{f,u} K[(g,m),(f,u)] X[(f,u),(b,j)]
__global__ void k_apply(const float* kbuf, int ldk, const float* x, int F, int dimIn,
                        int f0, int FC, float* out, int G, int dimOut, int B2) {
  int l = blockIdx.y;
  int d = 2 * l + 1, off = off_so3_d(l);
  int M = G * d, N = B2 * d, K = FC * d;
  int ntm = (M + TM - 1) / TM;
  if ((int)blockIdx.x >= ntm) return;
  LdK a{kbuf, ldk, d, G, off, M, K};
  LdX b{x, d, off, F, dimIn, f0, K, N};
  EpAcc ep{out, d, off, G, dimOut, M, N};
  gemm_core(blockIdx.x * TM, 0, K, a, b, ep);
}

// ------------------------------ pack / misc kernels ------------------------
struct Ptrs { const float* p[21]; };

__global__ void k_pack_dk(Ptrs s, float* dst) {
  int l = blockIdx.y, d = 2 * l + 1, dd = d * d, off = off_so3_d(l);
  int tot = NKER * dd;
  for (int i = blockIdx.x * blockDim.x + threadIdx.x; i < tot; i += gridDim.x * blockDim.x) {
    int n = i / dd, r = i - n * dd;
    dst[(size_t)n * DIM15 + off + r] = s.p[l][i];
  }
}
__global__ void k_pack_wig(Ptrs s, float* dst) {  // analysis: [1728, DIM20]
  int l = blockIdx.y, d = 2 * l + 1, dd = d * d, off = off_so3_d(l);
  float sc = sqrtf((float)(2 * l + 1));
  int tot = NGRID * dd;
  for (int i = blockIdx.x * blockDim.x + threadIdx.x; i < tot; i += gridDim.x * blockDim.x) {
    int n = i / dd, r = i - n * dd;
    dst[(size_t)n * DIM20 + off + r] = sc * s.p[l][i];
  }
}
__global__ void k_pack_wigT(Ptrs s, float* dst) {  // synthesis: [DIM15, 1728]
  int l = blockIdx.y, d = 2 * l + 1, dd = d * d, off = off_so3_d(l);
  float sc = sqrtf((float)(2 * l + 1));
  int tot = NGRID * dd;
  for (int i = blockIdx.x * blockDim.x + threadIdx.x; i < tot; i += gridDim.x * blockDim.x) {
    int n = i / dd, r = i - n * dd;
    dst[(size_t)(off + r) * NGRID + n] = sc * s.p[l][i];
  }
}
__global__ void k_pack_y(Ptrs s, float* dst) {  // [24, 441]
  int l = blockIdx.y, d = 2 * l + 1;
  int tot = NYG * d;
  for (int i = blockIdx.x * blockDim.x + threadIdx.x; i < tot; i += gridDim.x * blockDim.x) {
    int n = i / d, u = i - n * d;
    dst[(size_t)n * NSPH + l * l + u] = s.p[l][i];
  }
}
__global__ void k_zero(float* p, size_t n) {
  for (size_t i = blockIdx.x * (size_t)blockDim.x + threadIdx.x; i < n;
       i += (size_t)gridDim.x * blockDim.x)
    p[i] = 0.f;
}
// SO3->S2 apply (tiny: ~0.4 GFLOP): out[b,go, l^2+j] = sum_{f,u} K4[f*16+go, l^2+u] h[b,f,off+u*d+j]
__global__ void k_s2(const float* k4, const float* h, float* out) {
  int idx = blockIdx.x * blockDim.x + threadIdx.x;
  if (idx >= 2 * 16 * NSPH) return;
  int b = idx / (16 * NSPH), rem = idx - b * 16 * NSPH;
  int go = rem / NSPH, i = rem - go * NSPH;
  int l = 0;
  while ((l + 1) * (l + 1) <= i) ++l;
  int d = 2 * l + 1, j = i - l * l, off = off_so3_d(l);
  float acc = 0.f;
  for (int f = 0; f < 512; ++f) {
    const float* kp = k4 + (size_t)(f * 16 + go) * NSPH + l * l;
    const float* xp = h + ((size_t)b * 512 + f) * DIM20 + off + j;
    for (int u = 0; u < d; ++u) acc += kp[u] * xp[u * d];
  }
  out[idx] = acc;
}

// ------------------------------ host orchestration -------------------------
extern "C" void kernel_launch(void* const* d_in, const int* in_sizes, int n_in,
                              void* d_out, int out_size, void* d_ws, size_t ws_size,
                              hipStream_t stream) {
  (void)in_sizes; (void)out_size; (void)ws_size;
  const float* x  = (const float*)d_in[0];
  const float* w1 = (const float*)d_in[1];
  const float* w2 = (const float*)d_in[2];
  const float* w3 = (const float*)d_in[3];
  const float* w4 = (const float*)d_in[4];
  const bool flat = (n_in >= 64);  // tuples flattened to separate inputs, or concatenated
  Ptrs dk{}, dg{}, yy{};
  for (int l = 0; l < 16; ++l)
    dk.p[l] = flat ? (const float*)d_in[5 + l]
                   : (const float*)d_in[5] + (size_t)NKER * off_so3_h(l);
  for (int l = 0; l < 21; ++l)
    dg.p[l] = flat ? (const float*)d_in[21 + l]
                   : (const float*)d_in[6] + (size_t)NGRID * off_so3_h(l);
  const float* qw = flat ? (const float*)d_in[42] : (const float*)d_in[7];
  for (int l = 0; l < 21; ++l)
    yy.p[l] = flat ? (const float*)d_in[43 + l]
                   : (const float*)d_in[8] + (size_t)NYG * l * l;

  float* W = (float*)d_ws;
  size_t o = 0;
  auto take = [&](size_t nflt) { float* r = W + o; o += (nflt + 63) & ~(size_t)63; return r; };
  float* hA  = take((size_t)2 * 512 * DIM20);   // conv outputs (50.5 MB)
  float* hB  = take((size_t)2 * 512 * DIM20);   // activation outputs (50.5 MB)
  float* Sb  = take((size_t)2 * 512 * NGRID);   // grid signal (7 MB)
  float* DKF = take((size_t)NKER * DIM15);      // packed Dk (4.2 MB)
  float* WT  = take((size_t)DIM15 * NGRID);     // synthesis flat-Wigner^T (37.7 MB)
  float* WG  = take((size_t)NGRID * DIM20);     // analysis flat-Wigner (85.3 MB)
  float* KB  = take((size_t)4096 * DIM15);      // K streaming slice (89.4 MB, L2-resident)
  float* YF  = take((size_t)NYG * NSPH);        // packed Y

  // ---- pack static operands
  k_pack_dk  <<<dim3(128, 16), 256, 0, stream>>>(dk, DKF);
  k_pack_wig <<<dim3(256, 21), 256, 0, stream>>>(dg, WG);
  k_pack_wigT<<<dim3(256, 16), 256, 0, stream>>>(dg, WT);
  k_pack_y   <<<dim3(8, 21), 256, 0, stream>>>(yy, YF);

  const float sc1 = 1.f / sqrtf(192.f * 64.f);
  const float sc2 = 1.f / sqrtf(192.f * 128.f);
  const float sc3 = 1.f / sqrtf(192.f * 256.f);
  const float sc4 = 1.f / sqrtf(24.f * 512.f);

  // ---- conv1: x[2,64,1771] -> hA as h1[2,128,1771]
  k_zero<<<256, 256, 0, stream>>>(hA, (size_t)2 * 128 * DIM10);
  k_build<<<dim3(64 * 7, 11), 256, 0, stream>>>(w1, 8192, DKF, KB, DIM10, sc1);
  k_apply<<<dim3(21, 11), 256, 0, stream>>>(KB, DIM10, x, 64, DIM10, 0, 64, hA, 128, DIM10, 2);

  // ---- act1 (lmax 10): synth M=256 K=1771, analyze N=1771 -> hB
  k_gemm_rm<<<2 * 27, 256, 0, stream>>>(hA, DIM10, WT, NGRID, 256, NGRID, DIM10,
                                        EpSynth{Sb, qw, 256});
  k_gemm_rm<<<2 * 28, 256, 0, stream>>>(Sb, NGRID, WG, DIM20, 256, DIM10, NGRID,
                                        EpStore{hB, DIM10, 256, DIM10});

  // ---- conv2: hB[2,128,1771] -> hA as h2[2,256,5456] (l<=10 written, rest zero)
  k_zero<<<256, 256, 0, stream>>>(hA, (size_t)2 * 256 * DIM15);
  for (int fc = 0; fc < 4; ++fc) {
    k_build<<<dim3(64 * 7, 11), 256, 0, stream>>>(w2 + (size_t)fc * 32 * 256 * NKER, 8192,
                                                  DKF, KB, DIM10, sc2);
    k_apply<<<dim3(42, 11), 256, 0, stream>>>(KB, DIM10, hB, 128, DIM10, fc * 32, 32,
                                              hA, 256, DIM15, 2);
  }

  // ---- act2 (lmax 15): input nonzero only below DIM10 -> synth K=1771
  k_gemm_rm<<<4 * 27, 256, 0, stream>>>(hA, DIM15, WT, NGRID, 512, NGRID, DIM10,
                                        EpSynth{Sb, qw, 512});
  k_gemm_rm<<<4 * 86, 256, 0, stream>>>(Sb, NGRID, WG, DIM20, 512, DIM15, NGRID,
                                        EpStore{hB, DIM15, 512, DIM15});

  // ---- conv3: hB[2,256,5456] -> hA as h3[2,512,12341] (l<=15 written)
  k_zero<<<512, 256, 0, stream>>>(hA, (size_t)2 * 512 * DIM20);
  for (int fc = 0; fc < 32; ++fc) {
    k_build<<<dim3(32 * 16, 16), 256, 0, stream>>>(w3 + (size_t)fc * 8 * 512 * NKER, 4096,
                                                   DKF, KB, DIM15, sc3);
    k_apply<<<dim3(124, 16), 256, 0, stream>>>(KB, DIM15, hB, 256, DIM15, fc * 8, 8,
                                               hA, 512, DIM20, 2);
  }

  // ---- act3 (lmax 20): synth K=5456 (input zero beyond), analyze N=12341 -> hB
  k_gemm_rm<<<8 * 27, 256, 0, stream>>>(hA, DIM20, WT, NGRID, 1024, NGRID, DIM15,
                                        EpSynth{Sb, qw, 1024});
  k_gemm_rm<<<8 * 193, 256, 0, stream>>>(Sb, NGRID, WG, DIM20, 1024, DIM20, NGRID,
                                         EpStore{hB, DIM20, 1024, DIM20});

  // ---- SO3 -> S2: K4 = w4 @ Yflat (scaled), then small apply -> d_out [2,16,441]
  k_gemm_rm<<<64 * 7, 256, 0, stream>>>(w4, NYG, YF, NSPH, 8192, NSPH, NYG,
                                        EpScale{KB, NSPH, 8192, NSPH, sc4});
  k_s2<<<(2 * 16 * NSPH + 255) / 256, 256, 0, stream>>>(KB, hB, (float*)d_out);
}